// GNN_38147899523429
// MI455X (gfx1250) — compile-verified
//
#include <hip/hip_runtime.h>

// Problem constants (from reference)
#define NN 50000
#define EE 800000
#define BB 4
#define FF 64
#define RR (NN * BB)   // 200000 rows of 64 features, row r = n*4 + b

typedef float v2f __attribute__((ext_vector_type(2)));
typedef float v8f __attribute__((ext_vector_type(8)));

// ---------------- utility kernels ----------------

__global__ __launch_bounds__(256) void zero_f4_kernel(float4* p, int n4) {
    int t = blockIdx.x * 256 + threadIdx.x;
    if (t < n4) p[t] = make_float4(0.f, 0.f, 0.f, 0.f);
}

// x (B, N*64) batch-major  ->  h node-major: h[(n*4+b)*64 + f]
// exactly N*B*16 float4 elements; grid sized exactly (12500*256)
__global__ __launch_bounds__(256) void transpose_x_kernel(const float4* __restrict__ x4,
                                                          float4* __restrict__ h4) {
    int t  = blockIdx.x * 256 + threadIdx.x;  // t in [0, N*B*16)
    int f4 = t & 15;
    int b  = (t >> 4) & 3;
    int n  = t >> 6;
    h4[(size_t)(n * 4 + b) * 16 + f4] = x4[((size_t)b * NN + n) * 16 + f4];
}

__global__ __launch_bounds__(256) void deg_count_kernel(const int* __restrict__ dst,
                                                        float* __restrict__ deg) {
    int t = blockIdx.x * 256 + threadIdx.x;  // grid exact: E = 3125*256
    atomicAdd(&deg[dst[t]], 1.0f);
}

__global__ __launch_bounds__(256) void deg_inv_kernel(float* deg) {
    int t = blockIdx.x * 256 + threadIdx.x;
    if (t < NN) {
        float d = deg[t];
        deg[t] = (d > 0.f) ? (1.0f / d) : 0.0f;
    }
}

// Scatter-add of neighbor features. 64 threads per edge, each moves one float4
// (node block = 256 contiguous floats = all 4 batches). Grid exact: E*64 = 200000*256.
__global__ __launch_bounds__(256) void sage_agg_kernel(const float* __restrict__ h,
                                                       const int* __restrict__ src,
                                                       const int* __restrict__ dst,
                                                       float* __restrict__ agg) {
    int t    = blockIdx.x * 256 + threadIdx.x;
    int e    = t >> 6;
    int lane = t & 63;
    int s = src[e];
    int d = dst[e];
    float4 v = ((const float4*)(h + (size_t)s * 256))[lane];
    float* ap = agg + (size_t)d * 256 + lane * 4;
    atomicAdd(ap + 0, v.x);
    atomicAdd(ap + 1, v.y);
    atomicAdd(ap + 2, v.z);
    atomicAdd(ap + 3, v.w);
}

// ---------------- WMMA GEMM + bias + PReLU ----------------
// out[r,o] = sum_f (agg[r,f]*deg_inv[n]) * Wl[o,f] + bl[o] + sum_f h[r,f] * Wr[o,f]
// One wave per 16-row tile; 4 waves per block; 12500 tiles = 3125 blocks exactly
// (no partial waves -> EXEC all-1s for every WMMA, as the ISA requires).
__global__ __launch_bounds__(128) void sage_gemm_kernel(const float* __restrict__ agg,
                                                        const float* h,  // may alias out (rows private to wave)
                                                        const float* __restrict__ deg_inv,
                                                        const float* __restrict__ Wl,
                                                        const float* __restrict__ bl,
                                                        const float* __restrict__ Wr,
                                                        const float* __restrict__ aslope,
                                                        float* out, int bmajor_out) {
    // Stage transposed weights in LDS: sW[f*64 + o] = W[o*64 + f]  (B matrix is K x N = f x o)
    __shared__ float sWl[64 * 64];
    __shared__ float sWr[64 * 64];
    for (int i = threadIdx.x; i < 4096; i += 128) {
        int o = i >> 6, f = i & 63;
        sWl[f * 64 + o] = Wl[i];
        sWr[f * 64 + o] = Wr[i];
    }
    __syncthreads();

    int lane = threadIdx.x & 31;
    int wave = threadIdx.x >> 5;
    int tile = blockIdx.x * 4 + wave;   // always < 12500
    int m    = lane & 15;               // A: row within tile / B,D: column within 16-tile
    int half = lane >> 4;               // K-pair select for A/B fragments

    int r = tile * 16 + m;
    float ds = deg_inv[r >> 2];         // node index n = r / B
    const float* aRow = agg + (size_t)r * 64;
    const float* hRow = h + (size_t)r * 64;

    v8f acc[4];
    {
        v8f z = {};
#pragma unroll
        for (int i = 0; i < 4; ++i) acc[i] = z;
    }

#pragma unroll
    for (int kt = 0; kt < 16; ++kt) {
        int k = kt * 4 + half * 2;      // this lane's K pair within the 4-wide K step
        v2f am, ah;
        am.x = aRow[k] * ds;            // fused mean scaling
        am.y = aRow[k + 1] * ds;
        ah.x = hRow[k];
        ah.y = hRow[k + 1];
#pragma unroll
        for (int ot = 0; ot < 4; ++ot) {
            int o = ot * 16 + m;
            v2f bL, bR;
            bL.x = sWl[k * 64 + o];
            bL.y = sWl[(k + 1) * 64 + o];
            bR.x = sWr[k * 64 + o];
            bR.y = sWr[(k + 1) * 64 + o];
            acc[ot] = __builtin_amdgcn_wmma_f32_16x16x4_f32(
                false, am, false, bL, (short)0, acc[ot], false, false);
            acc[ot] = __builtin_amdgcn_wmma_f32_16x16x4_f32(
                false, ah, false, bR, (short)0, acc[ot], false, false);
        }
    }

    float slope = aslope[0];
    int rowb = tile * 16 + half * 8;    // D layout: M = v + 8*half
#pragma unroll
    for (int ot = 0; ot < 4; ++ot) {
        int o = ot * 16 + m;
        float bv = bl[o];
#pragma unroll
        for (int v = 0; v < 8; ++v) {
            float val = acc[ot][v] + bv;
            val = (val >= 0.f) ? val : slope * val;   // PReLU
            int row = rowb + v;
            if (!bmajor_out) {
                out[(size_t)row * 64 + o] = val;                 // node-major (in-place h)
            } else {
                int n = row >> 2, b = row & 3;
                out[((size_t)b * NN + n) * 64 + o] = val;        // final: batch-major d_out
            }
        }
    }
}

// ---------------- driver ----------------

extern "C" void kernel_launch(void* const* d_in, const int* in_sizes, int n_in,
                              void* d_out, int out_size, void* d_ws, size_t ws_size,
                              hipStream_t stream) {
    (void)in_sizes; (void)n_in; (void)out_size; (void)ws_size;

    const float* x  = (const float*)d_in[0];
    const int* ei   = (const int*)d_in[1];
    const int* src  = ei;         // edge_index[0,0,:]
    const int* dst  = ei + EE;    // edge_index[0,1,:]
    const float* Wl[3] = {(const float*)d_in[2], (const float*)d_in[6],  (const float*)d_in[10]};
    const float* bl[3] = {(const float*)d_in[3], (const float*)d_in[7],  (const float*)d_in[11]};
    const float* Wr[3] = {(const float*)d_in[4], (const float*)d_in[8],  (const float*)d_in[12]};
    const float* as[3] = {(const float*)d_in[5], (const float*)d_in[9],  (const float*)d_in[13]};

    // Workspace layout: [deg_inv: 50000 f] pad to 1 MiB | [agg: 12.8M f] | [h: 12.8M f]
    float* deg = (float*)d_ws;
    float* agg = (float*)((char*)d_ws + (1u << 20));
    float* hbf = agg + (size_t)RR * FF;

    // Node-major staging of x (51.2 MB -> L2-resident working set)
    transpose_x_kernel<<<12500, 256, 0, stream>>>((const float4*)x, (float4*)hbf);

    // Degree + inverse (shared graph across batches)
    zero_f4_kernel<<<49, 256, 0, stream>>>((float4*)deg, NN / 4);
    deg_count_kernel<<<EE / 256, 256, 0, stream>>>(dst, deg);
    deg_inv_kernel<<<196, 256, 0, stream>>>(deg);

    for (int L = 0; L < 3; ++L) {
        zero_f4_kernel<<<12500, 256, 0, stream>>>((float4*)agg, RR * FF / 4);
        sage_agg_kernel<<<(EE * 64) / 256, 256, 0, stream>>>(hbf, src, dst, agg);
        float* out = (L < 2) ? hbf : (float*)d_out;
        sage_gemm_kernel<<<3125, 128, 0, stream>>>(agg, hbf, deg,
                                                   Wl[L], bl[L], Wr[L], as[L],
                                                   out, (L == 2) ? 1 : 0);
    }
}